// NLLSequenceLoss_58832462021143
// MI455X (gfx1250) — compile-verified
//
#include <hip/hip_runtime.h>
#include <hip/hip_bf16.h>

// Problem constants (match reference)
#define NB 512
#define NT 128
#define NC 2000

typedef __attribute__((ext_vector_type(2))) float v2f;
typedef __attribute__((ext_vector_type(8))) float v8f;

// Kernel 1: gather the single relevant log-prob per batch row, negated.
// Only ~512 cachelines of the 524MB tensor are touched.
__global__ void nll_gather_kernel(const float* __restrict__ inputs,
                                  const int* __restrict__ length,
                                  const int* __restrict__ target,
                                  float* __restrict__ ws) {
    int b = blockIdx.x * blockDim.x + threadIdx.x;
    if (b < NB) {
        int len  = length[b];
        int last = (len < NT ? len : NT) - 1;          // min(T, length) - 1
        size_t idx = (size_t)b * (size_t)(NT * NC)
                   + (size_t)last * (size_t)NC
                   + (size_t)target[b];
        ws[b] = -inputs[idx];                           // loss at the masked step
    }
}

// Kernel 2: reduce 512 f32 values to their mean using FP32 WMMA.
// D = A(16x4) * ones(4x16) + C accumulates row sums of A into every column
// of the 16x16 f32 accumulator. 8 chained WMMAs consume all 512 values.
// Single wave32, EXEC all ones (WMMA requirement).
__global__ void nll_reduce_wmma_kernel(const float* __restrict__ ws,
                                       float* __restrict__ out) {
    const int lane = threadIdx.x;                       // 0..31

    v2f ones;
    ones.x = 1.0f;
    ones.y = 1.0f;

    v8f c = {};                                         // f32 accumulator (8 VGPRs)

#pragma unroll
    for (int chunk = 0; chunk < 8; ++chunk) {
        v2f a;
        // Any bijection of 64 values into the A tile sums identically;
        // use fully coalesced loads.
        a.x = ws[chunk * 64 + lane];
        a.y = ws[chunk * 64 + 32 + lane];
        // emits v_wmma_f32_16x16x4_f32 (accumulating chain)
        c = __builtin_amdgcn_wmma_f32_16x16x4_f32(
                /*neg_a=*/false, a,
                /*neg_b=*/false, ones,
                /*c_mod=*/(short)0, c,
                /*reuse_a=*/false, /*reuse_b=*/false);
    }

    // Per-lane: sum across the 8 accumulator VGPRs.
    // Lane l (l<16) holds sum over M=0..7 of column N=l; lane l+16 holds M=8..15.
    float local = c[0] + c[1] + c[2] + c[3] + c[4] + c[5] + c[6] + c[7];
    float other = __shfl_xor(local, 16, 32);            // pair halves of the wave

    if (lane == 0) {
        // mask.sum() == NB exactly (one valid timestep per row)
        out[0] = (local + other) * (1.0f / (float)NB);
    }
}

extern "C" void kernel_launch(void* const* d_in, const int* in_sizes, int n_in,
                              void* d_out, int out_size, void* d_ws, size_t ws_size,
                              hipStream_t stream) {
    const float* inputs = (const float*)d_in[0];   // [B, T, C] f32 log-probs
    const int*   length = (const int*)d_in[1];     // [B] int32
    const int*   target = (const int*)d_in[2];     // [B] int32
    float*       out    = (float*)d_out;           // scalar f32
    float*       ws     = (float*)d_ws;            // 512 f32 scratch

    nll_gather_kernel<<<dim3((NB + 255) / 256), dim3(256), 0, stream>>>(
        inputs, length, target, ws);

    nll_reduce_wmma_kernel<<<dim3(1), dim3(32), 0, stream>>>(ws, out);
}